// KGCompletionGNN_44616120271611
// MI455X (gfx1250) — compile-verified
//
#include <hip/hip_runtime.h>
#include <hip/hip_bf16.h>

#define DIM      128
#define TWO_D    256
#define NNODES   50000
#define NEDGES   600000
#define A_PITCH  132      // padded f32 pitch for 16-row A tiles (banks rotate by 4/row)
#define W_PITCH  130      // padded float2 pitch for weight-pair rows
#define WAVES    4

typedef __attribute__((ext_vector_type(2))) float v2f;
typedef __attribute__((ext_vector_type(8))) float v8f;

__device__ __forceinline__ float atomic_add_f32(float* p, float v) {
  return __hip_atomic_fetch_add(p, v, __ATOMIC_RELAXED, __HIP_MEMORY_SCOPE_AGENT);
}

// Stage W[:, koff:koff+128] transposed into LDS as k-pair float2s:
// sW[j*W_PITCH + n] = { W[n][koff+2j], W[n][koff+2j+1] }   (j=0..63, n=0..127)
__device__ __forceinline__ void stage_weights(const float* __restrict__ W, int koff,
                                              float2* sW, int tid, int nthreads) {
  for (int idx = tid; idx < 64 * DIM; idx += nthreads) {
    int j = idx & 63;
    int n = idx >> 6;
    const float* wp = W + n * TWO_D + koff + 2 * j;
    float2 p;
    p.x = wp[0];
    p.y = wp[1];
    sW[j * W_PITCH + n] = p;
  }
}

// Async-stage 16 consecutive rows of X (row length 128 f32) into padded LDS A tile.
// One GLOBAL_LOAD_ASYNC_TO_LDS_B128 per row: 32 lanes x 16B = 512B = full row,
// global -> LDS with no VGPR round-trip, tracked on ASYNCcnt.
__device__ __forceinline__ void stage_rows_async(const float* __restrict__ X, long rowbase,
                                                 float* sA, int lane) {
#pragma unroll
  for (int m = 0; m < 16; ++m) {
    const float* gp = X + (rowbase + m) * (long)DIM + lane * 4;
    unsigned lds = (unsigned)(uintptr_t)(sA + m * A_PITCH + lane * 4);
    asm volatile("global_load_async_to_lds_b128 %0, %1, off"
                 :: "v"(lds), "v"(gp) : "memory");
  }
}

__device__ __forceinline__ void wait_async_le16() {
  asm volatile("s_wait_asynccnt 0x10" ::: "memory");
}
__device__ __forceinline__ void wait_async_zero() {
  asm volatile("s_wait_asynccnt 0x0" ::: "memory");
}

// 16x128 f32 GEMM tile via V_WMMA_F32_16X16X4_F32:
//   acc[n] (16x16, n-th column tile) += A(16x128 from sA) x Wt(128x128 from sW)
__device__ __forceinline__ void wmma_16x128(const float* sA, const float2* sW,
                                            int lane, v8f acc[8]) {
  const int lo = lane & 15;
  const int hi = lane >> 4;  // 0 or 1
#pragma unroll
  for (int ks = 0; ks < 32; ++ks) {     // K step of 4
    // A frag: element (m=lo, k = 4*ks + 2*hi + v) -> two consecutive f32
    const float* ap = sA + lo * A_PITCH + ks * 4 + 2 * hi;
    v2f a;
    a.x = ap[0];
    a.y = ap[1];
    const float2* bp = sW + (ks * 2 + hi) * W_PITCH + lo;
#pragma unroll
    for (int n = 0; n < 8; ++n) {
      // B frag: element (k = 4*ks + 2*hi + v, col = n*16+lo)
      float2 bw = bp[n * 16];
      v2f b;
      b.x = bw.x;
      b.y = bw.y;
      acc[n] = __builtin_amdgcn_wmma_f32_16x16x4_f32(false, a, false, b,
                                                     (short)0, acc[n], false, false);
    }
  }
}

// ---------------- Kernel 0: zero agg + counts ----------------
__global__ void zero2_kernel(float* a, size_t na, float* b, size_t nb) {
  size_t i = (size_t)blockIdx.x * blockDim.x + threadIdx.x;
  size_t stride = (size_t)gridDim.x * blockDim.x;
  for (size_t k = i; k < na; k += stride) a[k] = 0.0f;
  for (size_t k = i; k < nb; k += stride) b[k] = 0.0f;
}

// ---------------- Kernel 1: HW = H @ W[:, :128].T + bias  (grid.y: 0=fwd,1=back) ----------------
__global__ __launch_bounds__(128) void hw_rows_kernel(
    const float* __restrict__ H,
    const float* __restrict__ W_fwd, const float* __restrict__ b_fwd,
    const float* __restrict__ W_back, const float* __restrict__ b_back,
    float* __restrict__ HWf, float* __restrict__ HWb) {
  __shared__ alignas(16) float2 sW[64 * W_PITCH];
  __shared__ alignas(16) float sA[WAVES][2][16 * A_PITCH];
  const int tid = threadIdx.x;
  const int lane = tid & 31;
  const int wave = tid >> 5;
  const int pass = blockIdx.y;
  const float* W = pass ? W_back : W_fwd;
  const float* bias = pass ? b_back : b_fwd;
  float* out = pass ? HWb : HWf;

  stage_weights(W, 0, sW, tid, blockDim.x);
  __syncthreads();

  const int lo = lane & 15, hi = lane >> 4;
  float bcol[8];
#pragma unroll
  for (int n = 0; n < 8; ++n) bcol[n] = bias[n * 16 + lo];

  const int ntiles = NNODES / 16;  // 3125
  const int stride = gridDim.x * WAVES;
  int t = blockIdx.x * WAVES + wave;
  int cur = 0;
  if (t < ntiles) stage_rows_async(H, (long)t * 16, sA[wave][0], lane);
  for (; t < ntiles; t += stride) {
    int tn = t + stride;
    if (tn < ntiles) {
      stage_rows_async(H, (long)tn * 16, sA[wave][cur ^ 1], lane);
      wait_async_le16();   // oldest 16 (current buffer) complete
    } else {
      wait_async_zero();
    }
    long rowbase = (long)t * 16;
    v8f acc[8] = {};
    wmma_16x128(sA[wave][cur], sW, lane, acc);
#pragma unroll
    for (int r = 0; r < 8; ++r) {
      long row = rowbase + r + 8 * hi;
#pragma unroll
      for (int n = 0; n < 8; ++n)
        out[row * DIM + n * 16 + lo] = acc[n][r] + bcol[n];
    }
    cur ^= 1;
  }
}

// ---------------- Kernel 2: edge GEMM (E part) + gather HW + scatter-add ----------------
// pass 0 (fwd):  agg[t] += E@Wf[:,128:].T + HWf[h];  cnt[t] += 1
// pass 1 (back): agg[h] += E@Wb[:,128:].T + HWb[t];  cnt[h] += 1
__global__ __launch_bounds__(128) void edge_kernel(
    const float* __restrict__ E, const int* __restrict__ ht,
    const float* __restrict__ W_fwd, const float* __restrict__ W_back,
    const float* __restrict__ HWf, const float* __restrict__ HWb,
    float* __restrict__ agg, float* __restrict__ cnt) {
  __shared__ alignas(16) float2 sW[64 * W_PITCH];
  __shared__ alignas(16) float sA[WAVES][2][16 * A_PITCH];
  __shared__ int sIdx[WAVES][32];
  const int tid = threadIdx.x;
  const int lane = tid & 31;
  const int wave = tid >> 5;
  const int pass = blockIdx.y;
  const float* W = pass ? W_back : W_fwd;
  const float* HWsrc = pass ? HWb : HWf;

  stage_weights(W, DIM, sW, tid, blockDim.x);
  __syncthreads();

  const int lo = lane & 15, hi = lane >> 4;
  const int ntiles = NEDGES / 16;  // 37500
  const int stride = gridDim.x * WAVES;
  int t = blockIdx.x * WAVES + wave;
  int cur = 0;
  if (t < ntiles) stage_rows_async(E, (long)t * 16, sA[wave][0], lane);
  for (; t < ntiles; t += stride) {
    int tn = t + stride;
    if (tn < ntiles) {
      stage_rows_async(E, (long)tn * 16, sA[wave][cur ^ 1], lane);
      wait_async_le16();   // oldest 16 (current buffer) complete
    } else {
      wait_async_zero();
    }
    long ebase = (long)t * 16;
    if (lane < 16) {
      int e2 = 2 * (int)(ebase + lane);
      int h = ht[e2];
      int tt = ht[e2 + 1];
      sIdx[wave][lane] = pass ? h : tt;        // scatter destination
      sIdx[wave][lane + 16] = pass ? tt : h;   // gather source
    }
    v8f acc[8] = {};
    wmma_16x128(sA[wave][cur], sW, lane, acc);

    if (lane < 16) atomic_add_f32(&cnt[sIdx[wave][lane]], 1.0f);

#pragma unroll
    for (int r = 0; r < 8; ++r) {
      int m = r + 8 * hi;
      long dst = sIdx[wave][m];
      long g = sIdx[wave][16 + m];
      const float* hwRow = HWsrc + g * DIM;
      float* aggRow = agg + dst * DIM;
#pragma unroll
      for (int n = 0; n < 8; ++n) {
        int col = n * 16 + lo;
        atomic_add_f32(aggRow + col, acc[n][r] + hwRow[col]);
      }
    }
    cur ^= 1;
  }
}

// ---------------- Kernel 3: mean-divide, leaky_relu, residual, LayerNorm ----------------
__global__ __launch_bounds__(256) void finalize_kernel(
    const float* __restrict__ agg, const float* __restrict__ cnt,
    const float* __restrict__ H, const float* __restrict__ scale,
    const float* __restrict__ bias, float* __restrict__ out) {
  int node = blockIdx.x * 8 + (threadIdx.x >> 5);  // one wave32 per node
  int lane = threadIdx.x & 31;
  if (node >= NNODES) return;
  long base = (long)node * DIM + lane * 4;
  float4 a = *(const float4*)(agg + base);
  float4 h = *(const float4*)(H + base);
  float inv = 1.0f / (cnt[node] + 1e-7f);
  float x[4];
  {
    float z0 = a.x * inv, z1 = a.y * inv, z2 = a.z * inv, z3 = a.w * inv;
    x[0] = (z0 > 0.0f ? z0 : 0.01f * z0) + h.x;
    x[1] = (z1 > 0.0f ? z1 : 0.01f * z1) + h.y;
    x[2] = (z2 > 0.0f ? z2 : 0.01f * z2) + h.z;
    x[3] = (z3 > 0.0f ? z3 : 0.01f * z3) + h.w;
  }
  float s = x[0] + x[1] + x[2] + x[3];
  float ss = x[0] * x[0] + x[1] * x[1] + x[2] * x[2] + x[3] * x[3];
#pragma unroll
  for (int off = 16; off > 0; off >>= 1) {
    s += __shfl_xor(s, off, 32);
    ss += __shfl_xor(ss, off, 32);
  }
  float mean = s * (1.0f / 128.0f);
  float var = ss * (1.0f / 128.0f) - mean * mean;
  float rstd = rsqrtf(var + 1e-5f);
  float4 sc = *(const float4*)(scale + lane * 4);
  float4 bi = *(const float4*)(bias + lane * 4);
  float4 o;
  o.x = (x[0] - mean) * rstd * sc.x + bi.x;
  o.y = (x[1] - mean) * rstd * sc.y + bi.y;
  o.z = (x[2] - mean) * rstd * sc.z + bi.z;
  o.w = (x[3] - mean) * rstd * sc.w + bi.w;
  *(float4*)(out + base) = o;
}

extern "C" void kernel_launch(void* const* d_in, const int* in_sizes, int n_in,
                              void* d_out, int out_size, void* d_ws, size_t ws_size,
                              hipStream_t stream) {
  const float* H = (const float*)d_in[0];
  const float* E = (const float*)d_in[1];
  const int* ht = (const int*)d_in[2];
  const float* W_fwd = (const float*)d_in[3];
  const float* b_fwd = (const float*)d_in[4];
  const float* W_back = (const float*)d_in[5];
  const float* b_back = (const float*)d_in[6];
  const float* ln_scale = (const float*)d_in[7];
  const float* ln_bias = (const float*)d_in[8];
  float* out = (float*)d_out;

  // workspace layout (floats): agg[50000*128] | HWf[50000*128] | HWb[50000*128] | cnt[50000]
  float* ws = (float*)d_ws;
  float* agg = ws;
  float* HWf = agg + (size_t)NNODES * DIM;
  float* HWb = HWf + (size_t)NNODES * DIM;
  float* cnt = HWb + (size_t)NNODES * DIM;

  zero2_kernel<<<2048, 256, 0, stream>>>(agg, (size_t)NNODES * DIM, cnt, (size_t)NNODES);
  hw_rows_kernel<<<dim3(256, 2), 128, 0, stream>>>(H, W_fwd, b_fwd, W_back, b_back, HWf, HWb);
  edge_kernel<<<dim3(1024, 2), 128, 0, stream>>>(E, ht, W_fwd, W_back, HWf, HWb, agg, cnt);
  finalize_kernel<<<NNODES / 8, 256, 0, stream>>>(agg, cnt, H, ln_scale, ln_bias, out);
}